// WassersteinLoss_20186346291333
// MI455X (gfx1250) — compile-verified
//
#include <hip/hip_runtime.h>
#include <hip/hip_bf16.h>

// ---------------------------------------------------------------------------
// Bures-Wasserstein distance between 2x2 SPD blocks of 3x3 inputs (B pairs).
// Memory-bound (76 B / ~25 FLOP per element) -> optimize data movement only.
// gfx1250 path: async global->LDS staging (ASYNCcnt) for coalesced AoS reads.
// Full tiles take an unguarded staging/compute path (no EXEC-mask churn).
// ---------------------------------------------------------------------------

#define BW_EPS 1e-7f
#define TILE 512                      // elements per workgroup tile
#define THREADS 256                   // 8 waves (wave32)
#define REC 9                         // floats per 3x3 record
#define TILE_FLOATS (TILE * REC)      // 4608 floats = 18432 bytes per input
#define TILE_CHUNKS (TILE_FLOATS / 4) // 1152 x 16-byte async chunks
#define FULL_ROUNDS (TILE_CHUNKS / THREADS)        // 4 uniform rounds
#define REM_CHUNKS  (TILE_CHUNKS - FULL_ROUNDS * THREADS) // 128 leftover

typedef int v4i_t __attribute__((vector_size(16)));  // 128-bit chunk type

// --- gfx1250 async global->LDS b128: builtin if available, else inline asm ---
#if __has_builtin(__builtin_amdgcn_global_load_async_to_lds_b128)
#define ASYNC_LOAD_B128(gp, lp)                                              \
  __builtin_amdgcn_global_load_async_to_lds_b128(                            \
      (__attribute__((address_space(1))) v4i_t*)(void*)(gp),                 \
      (__attribute__((address_space(3))) v4i_t*)(void*)(lp), 0, 0)
#else
#define ASYNC_LOAD_B128(gp, lp)                                              \
  asm volatile("global_load_async_to_lds_b128 %0, %1, off"                   \
               :: "v"((unsigned)(size_t)(__attribute__((address_space(3)))   \
                                         void*)(void*)(lp)),                 \
                  "v"((unsigned long long)(size_t)(gp))                      \
               : "memory")
#endif

__device__ __forceinline__ void wait_async_zero() {
#if __has_builtin(__builtin_amdgcn_s_wait_asynccnt)
  __builtin_amdgcn_s_wait_asynccnt(0);
#else
  asm volatile("s_wait_asynccnt 0x0" ::: "memory");
#endif
}

// Closed-form 2x2 Bures-Wasserstein from LDS-resident records.
// Record layout (row-major 3x3): a=+0, b=+1 (==+3 by symmetry), c=+4.
__device__ __forceinline__ float bw2x2(const float* __restrict__ r1,
                                       const float* __restrict__ r2) {
  const float a1 = r1[0], b1 = r1[1], c1 = r1[4];
  const float a2 = r2[0], b2 = r2[1], c2 = r2[4];
  // t = tr(C1*C2), d = det(C1)*det(C2); eigs of inner == eigs of C1*C2
  const float t  = fmaf(a1, a2, fmaf(2.0f * b1, b2, c1 * c2));
  const float d1 = fmaf(a1, c1, -b1 * b1);
  const float d2 = fmaf(a2, c2, -b2 * b2);
  const float disc = sqrtf(fmaxf(fmaf(t, t, -4.0f * (d1 * d2)), 0.0f));
  const float l1 = fmaxf(0.5f * (t + disc), BW_EPS);
  const float l2 = fmaxf(0.5f * (t - disc), BW_EPS);
  const float tr_sqrt_inner = sqrtf(l1) + sqrtf(l2);
  float w = (a1 + c1) + (a2 + c2) - 2.0f * tr_sqrt_inner;
  if (w != w) w = 10.0f;               // nan_to_num(nan=10.0)
  return w;
}

__global__ __launch_bounds__(THREADS)
void bures_wasserstein_2x2_kernel(const float* __restrict__ A1,
                                  const float* __restrict__ A2,
                                  float* __restrict__ out, int B) {
  __shared__ float lds1[TILE_FLOATS];
  __shared__ float lds2[TILE_FLOATS];

  const int tid = threadIdx.x;
  const long tileBase = (long)blockIdx.x * TILE;   // first element of tile
  const long fBase = tileBase * REC;               // float offset into A1/A2

  if (tileBase + TILE <= (long)B) {
    // ================= fast path: full tile, no guards =================
    const float* __restrict__ g1 = A1 + fBase;
    const float* __restrict__ g2 = A2 + fBase;
#pragma unroll
    for (int j = 0; j < FULL_ROUNDS; ++j) {
      const int c = j * THREADS + tid;             // uniform, fully coalesced
      ASYNC_LOAD_B128(g1 + c * 4, &lds1[c * 4]);
      ASYNC_LOAD_B128(g2 + c * 4, &lds2[c * 4]);
    }
    if (tid < REM_CHUNKS) {                        // chunks 1024..1151
      const int c = FULL_ROUNDS * THREADS + tid;
      ASYNC_LOAD_B128(g1 + c * 4, &lds1[c * 4]);
      ASYNC_LOAD_B128(g2 + c * 4, &lds2[c * 4]);
    }
    wait_async_zero();   // this wave's async copies landed in LDS
    __syncthreads();     // all waves' copies visible

#pragma unroll
    for (int k = 0; k < TILE / THREADS; ++k) {
      const int e = k * THREADS + tid;
      out[tileBase + e] = bw2x2(&lds1[e * REC], &lds2[e * REC]);
    }
  } else {
    // ================= tail path: ragged tile, guarded =================
    const long totalFloats = (long)B * REC;
    for (int c = tid; c < TILE_CHUNKS; c += THREADS) {
      const long f = fBase + (long)c * 4;
      if (f < totalFloats) {
        ASYNC_LOAD_B128(A1 + f, &lds1[c * 4]);
        ASYNC_LOAD_B128(A2 + f, &lds2[c * 4]);
      }
    }
    wait_async_zero();
    __syncthreads();

#pragma unroll
    for (int k = 0; k < TILE / THREADS; ++k) {
      const int e = k * THREADS + tid;
      const long idx = tileBase + e;
      if (idx < B) out[idx] = bw2x2(&lds1[e * REC], &lds2[e * REC]);
    }
  }
}

extern "C" void kernel_launch(void* const* d_in, const int* in_sizes, int n_in,
                              void* d_out, int out_size, void* d_ws, size_t ws_size,
                              hipStream_t stream) {
  (void)n_in; (void)out_size; (void)d_ws; (void)ws_size;
  const float* A1 = (const float*)d_in[0];
  const float* A2 = (const float*)d_in[1];
  float* out = (float*)d_out;
  const int B = in_sizes[0] / (REC);       // elements (3x3 records)
  if (B <= 0) return;
  const int nTiles = (B + TILE - 1) / TILE;
  bures_wasserstein_2x2_kernel<<<nTiles, THREADS, 0, stream>>>(A1, A2, out, B);
}